// GaussianMixture_57131654972098
// MI455X (gfx1250) — compile-verified
//
#include <hip/hip_runtime.h>
#include <math.h>

typedef __attribute__((ext_vector_type(2))) float v2f;
typedef __attribute__((ext_vector_type(8))) float v8f;

#define MCOMP 2048
#define CSTRIDE 8   // 6 coefficients + 2 zero pad -> K=8 (two K=4 WMMA steps)

// ---------------- Kernel 1: lse = logsumexp(w) over M components ----------------
__global__ void gmm_lse_w(const float* __restrict__ w, float* __restrict__ ws, int M) {
    __shared__ float red[256];
    int tid = threadIdx.x;
    float mx = -INFINITY;
    for (int i = tid; i < M; i += 256) mx = fmaxf(mx, w[i]);
    red[tid] = mx; __syncthreads();
    for (int s = 128; s > 0; s >>= 1) {
        if (tid < s) red[tid] = fmaxf(red[tid], red[tid + s]);
        __syncthreads();
    }
    float gmax = red[0]; __syncthreads();
    float sum = 0.f;
    for (int i = tid; i < M; i += 256) sum += __expf(w[i] - gmax);
    red[tid] = sum; __syncthreads();
    for (int s = 128; s > 0; s >>= 1) {
        if (tid < s) red[tid] += red[tid + s];
        __syncthreads();
    }
    if (tid == 0) ws[MCOMP * CSTRIDE] = gmax + logf(red[0]);
}

// ---------------- Kernel 2: per-component coefficient vectors (rank-6 factorization) ----
__global__ void gmm_coef(const float* __restrict__ mu, const float* __restrict__ A,
                         const float* __restrict__ w, float* __restrict__ ws, int M) {
    int m = blockIdx.x * blockDim.x + threadIdx.x;
    if (m >= M) return;
    float lse = ws[MCOMP * CSTRIDE];
    float a00 = A[m * 4 + 0], a01 = A[m * 4 + 1];
    float a10 = A[m * 4 + 2], a11 = A[m * 4 + 3];
    // G = 0.5 * A A^T  (symmetric -> g01 == g10)
    float g00 = 0.5f * (a00 * a00 + a01 * a01);
    float g01 = 0.5f * (a00 * a10 + a01 * a11);
    float g11 = 0.5f * (a10 * a10 + a11 * a11);
    float det = g00 * g11 - g01 * g01;
    float wlog = w[m] - lse + 0.5f * logf(det);
    float mu0 = mu[m * 2 + 0], mu1 = mu[m * 2 + 1];
    float c01 = 2.f * g01;   // (g01 + g10)
    float* cf = ws + m * CSTRIDE;
    cf[0] = -g00;
    cf[1] = -c01;
    cf[2] = -g11;
    cf[3] = 2.f * g00 * mu0 + c01 * mu1;
    cf[4] = 2.f * g11 * mu1 + c01 * mu0;
    cf[5] = wlog - (g00 * mu0 * mu0 + c01 * mu0 * mu1 + g11 * mu1 * mu1);
    cf[6] = 0.f;
    cf[7] = 0.f;
}

// ---------------- Kernel 3: WMMA GEMM + fused logsumexp ----------------
// Each wave (32 lanes) owns a tile of 16 samples; loops over all 2048 components
// in 16-wide chunks using V_WMMA_F32_16X16X4_F32 (two chained for K=8).
__global__ __launch_bounds__(256) void gmm_ll(const float* __restrict__ sample,
                                              const float* __restrict__ coef,
                                              float* __restrict__ out, int nTiles) {
    __shared__ float lds[MCOMP * CSTRIDE];   // 64 KB coefficient table
    // Cooperative stage of coefficients into LDS (float4 = 16B per load)
    {
        const float4* src = (const float4*)coef;
        float4* dst = (float4*)lds;
        for (int i = threadIdx.x; i < (MCOMP * CSTRIDE) / 4; i += blockDim.x) dst[i] = src[i];
    }
    __syncthreads();

    const int lane = threadIdx.x & 31;
    const int wave = threadIdx.x >> 5;
    const int tile = blockIdx.x * (blockDim.x >> 5) + wave;
    if (tile >= nTiles) return;            // wave-uniform branch

    const int base = tile * 16;
    const int mrow = lane & 15;            // sample row / component column owned by lane
    const int half = lane >> 4;            // K-half select per ISA A/B layout
    const int k0 = half * 2;

    const float x0 = sample[(base + mrow) * 2 + 0];
    const float x1 = sample[(base + mrow) * 2 + 1];
    // Feature vector f = [x0^2, x0*x1, x1^2, x0, x1, 1, 0, 0]
    // A-matrix layout (16x4 f32): lane l holds row l&15; K={0,1} for l<16, K={2,3} for l>=16
    v2f a0, a1;
    a0.x = half ? (x1 * x1) : (x0 * x0);
    a0.y = half ? x0        : (x0 * x1);
    a1.x = half ? 0.f       : x1;
    a1.y = half ? 0.f       : 1.f;

    float mx[8];
#pragma unroll
    for (int j = 0; j < 8; ++j) mx[j] = -INFINITY;

    const int nCh = MCOMP / 16;

    // ---- Pass 1: max of logits per sample row ----
    for (int ch = 0; ch < nCh; ++ch) {
        const float* cf = &lds[(ch * 16 + mrow) * CSTRIDE];
        v2f b0, b1;
        b0.x = cf[k0];     b0.y = cf[k0 + 1];
        b1.x = cf[4 + k0]; b1.y = cf[5 + k0];
        v8f c = {};
        c = __builtin_amdgcn_wmma_f32_16x16x4_f32(false, a0, false, b0, (short)0, c, false, false);
        c = __builtin_amdgcn_wmma_f32_16x16x4_f32(false, a1, false, b1, (short)0, c, false, false);
#pragma unroll
        for (int j = 0; j < 8; ++j) mx[j] = fmaxf(mx[j], c[j]);
    }
    // Reduce max across the 16-lane group (halves never mix: masks < 16)
#pragma unroll
    for (int off = 8; off >= 1; off >>= 1)
#pragma unroll
        for (int j = 0; j < 8; ++j) mx[j] = fmaxf(mx[j], __shfl_xor(mx[j], off, 32));

    // ---- Pass 2: sum of exp(logit - max) ----
    float sum[8];
#pragma unroll
    for (int j = 0; j < 8; ++j) sum[j] = 0.f;

    for (int ch = 0; ch < nCh; ++ch) {
        const float* cf = &lds[(ch * 16 + mrow) * CSTRIDE];
        v2f b0, b1;
        b0.x = cf[k0];     b0.y = cf[k0 + 1];
        b1.x = cf[4 + k0]; b1.y = cf[5 + k0];
        v8f c = {};
        c = __builtin_amdgcn_wmma_f32_16x16x4_f32(false, a0, false, b0, (short)0, c, false, false);
        c = __builtin_amdgcn_wmma_f32_16x16x4_f32(false, a1, false, b1, (short)0, c, false, false);
#pragma unroll
        for (int j = 0; j < 8; ++j) sum[j] += __expf(c[j] - mx[j]);
    }
#pragma unroll
    for (int off = 8; off >= 1; off >>= 1)
#pragma unroll
        for (int j = 0; j < 8; ++j) sum[j] += __shfl_xor(sum[j], off, 32);

    // D layout: VGPR j = row j (lanes 0-15) / row j+8 (lanes 16-31); lane 0 & 16 write
    if (mrow == 0) {
#pragma unroll
        for (int j = 0; j < 8; ++j) out[base + half * 8 + j] = mx[j] + logf(sum[j]);
    }
}

extern "C" void kernel_launch(void* const* d_in, const int* in_sizes, int n_in,
                              void* d_out, int out_size, void* d_ws, size_t ws_size,
                              hipStream_t stream) {
    const float* sample = (const float*)d_in[0];   // (N, 2) f32
    const float* mu     = (const float*)d_in[1];   // (M, 2) f32
    const float* A      = (const float*)d_in[2];   // (M, 2, 2) f32
    const float* w      = (const float*)d_in[3];   // (M, 1) f32
    float* out = (float*)d_out;                    // (N, 1) f32
    float* ws  = (float*)d_ws;                     // [0, M*8) coef table, [M*8] = lse(w)

    const int N = in_sizes[0] / 2;
    const int M = in_sizes[3];                     // == MCOMP

    gmm_lse_w<<<1, 256, 0, stream>>>(w, ws, M);
    gmm_coef<<<(M + 255) / 256, 256, 0, stream>>>(mu, A, w, ws, M);

    const int nTiles = N / 16;                     // N = 131072 -> 8192 tiles
    const int wavesPerBlock = 8;                   // 256 threads, wave32
    const int nBlocks = (nTiles + wavesPerBlock - 1) / wavesPerBlock;
    gmm_ll<<<nBlocks, 256, 0, stream>>>(sample, ws, out, nTiles);
}